// GBDT2NN_10067403342067
// MI455X (gfx1250) — compile-verified
//
#include <hip/hip_runtime.h>
#include <cstddef>
#include <cstdint>

typedef __attribute__((ext_vector_type(2))) float v2f;
typedef __attribute__((ext_vector_type(8))) float v8f;

constexpr int kB  = 8192;   // batch
constexpr int kIN = 1024;   // input features
constexpr int kG  = 100;    // groups (models)
constexpr int kF  = 128;    // features per group
constexpr int kL0 = 100;
constexpr int kL1 = 100;
constexpr int kL2 = 50;
constexpr int kWaves   = 8;                    // waves per block (wave32)
constexpr int kMBlocks = kB / (16 * kWaves);   // 64 row-blocks of 128 rows
constexpr int kNT01 = 7;                       // n-tiles for 100 cols
constexpr int kNT2  = 4;                       // n-tiles for 50 cols
constexpr float kInvB = 1.0f / (float)kB;
constexpr float kEps  = 1e-5f;

__device__ __forceinline__ v8f wmma_f32(v2f a, v2f b, v8f c) {
  // D = A(16x4) * B(4x16) + C(16x16), fp32 matrix pipe
  return __builtin_amdgcn_wmma_f32_16x16x4_f32(
      /*neg_a=*/false, a, /*neg_b=*/false, b,
      /*c_mod=*/(short)0, c, /*reuse_a=*/false, /*reuse_b=*/false);
}

// ---------------------------------------------------------------------------
// Layer 0: gather A once into registers, full group weights in LDS,
// loop all 7 n-tiles. Store pre-BN Z0 + deterministic per-block stats.
// ---------------------------------------------------------------------------
__global__ __launch_bounds__(256) void gbdt_gemm0(
    const float* __restrict__ x, const int* __restrict__ used,
    const float* __restrict__ w0, const float* __restrict__ b0,
    float* __restrict__ Z0, float* __restrict__ psum, float* __restrict__ psq)
{
  __shared__ alignas(16) int   uf[kF];
  __shared__ alignas(16) float wtT[kNT01 * 16][kF + 4];   // [n][k], 112x132
  __shared__ float bias_s[kNT01 * 16];
  __shared__ float wsum[kWaves][16];
  __shared__ float wsq[kWaves][16];

  const int g    = blockIdx.y;
  const int tid  = threadIdx.x;
  const int wave = tid >> 5;
  const int lane = tid & 31;
  const int half = lane >> 4;   // K-half select (A/B fragment layout)
  const int r    = lane & 15;   // M for A-frag, N for B/C-frag
  const int m0   = (blockIdx.x * kWaves + wave) * 16;

  for (int i = tid; i < kF; i += 256) uf[i] = used[g * kF + i];
  for (int i = tid; i < kF * kNT01 * 16; i += 256) {
    const int k = i / (kNT01 * 16), n = i % (kNT01 * 16);
    wtT[n][k] = (n < kL0) ? w0[((size_t)g * kF + k) * kL0 + n] : 0.0f;
  }
  if (tid < kNT01 * 16) bias_s[tid] = (tid < kL0) ? b0[g * kL0 + tid] : 0.0f;
  __syncthreads();

  // A fragments for 16 rows x 128 k, held in registers (64 VGPRs/lane).
  // All loops touching afrag are FULLY unrolled -> constant indices only,
  // so the array stays in VGPRs (no scratch spill).
  const float* __restrict__ xrow = x + (size_t)(m0 + r) * kIN;
  v2f afrag[kF / 4];
#pragma unroll
  for (int kb = 0; kb < kF / 4; ++kb) {
    const int  k   = kb * 4 + half * 2;
    const int2 idx = *(const int2*)&uf[k];
    afrag[kb].x = xrow[idx.x];
    afrag[kb].y = xrow[idx.y];
  }

  for (int nt = 0; nt < kNT01; ++nt) {
    const int n0 = nt * 16;
    v8f acc = {};
#pragma unroll
    for (int kb = 0; kb < kF / 4; ++kb) {
      const int k = kb * 4 + half * 2;
      const v2f b = *(const v2f*)&wtT[n0 + r][k];
      acc = wmma_f32(afrag[kb], b, acc);
    }
    const int n = n0 + r;
    float s = 0.0f, q = 0.0f;
    if (n < kL0) {
      const float bia = bias_s[n];
      float* __restrict__ zp = Z0 + ((size_t)g * kB + m0 + half * 8) * kL0 + n;
#pragma unroll
      for (int j = 0; j < 8; ++j) {
        const float z = acc[j] + bia;   // C/D layout: row = j + 8*half, col = r
        zp[j * kL0] = z;
        s += z;
        q += z * z;
      }
    }
    s += __shfl_xor(s, 16);
    q += __shfl_xor(q, 16);
    if (half == 0) { wsum[wave][r] = s; wsq[wave][r] = q; }
    __syncthreads();
    if (tid < 16 && (n0 + tid) < kL0) {
      float ts = 0.0f, tq = 0.0f;
#pragma unroll
      for (int w = 0; w < kWaves; ++w) { ts += wsum[w][tid]; tq += wsq[w][tid]; }
      const size_t c = (size_t)(g * kL0 + n0 + tid) * kMBlocks + blockIdx.x;
      psum[c] = ts;
      psq[c]  = tq;
    }
    __syncthreads();
  }
}

// ---------------------------------------------------------------------------
// Reduce per-block partials -> BN scale/shift (fixed order, deterministic).
// ---------------------------------------------------------------------------
__global__ __launch_bounds__(256) void gbdt_bnprep(
    const float* __restrict__ psum, const float* __restrict__ psq,
    const float* __restrict__ gamma, const float* __restrict__ beta,
    float* __restrict__ scale, float* __restrict__ shift)
{
  const int c = blockIdx.x * 256 + threadIdx.x;
  if (c >= kG * kL0) return;
  float s = 0.0f, q = 0.0f;
  for (int p = 0; p < kMBlocks; ++p) {
    s += psum[(size_t)c * kMBlocks + p];
    q += psq[(size_t)c * kMBlocks + p];
  }
  const float mu  = s * kInvB;
  const float var = q * kInvB - mu * mu;
  const float sc  = gamma[c] * rsqrtf(var + kEps);
  scale[c] = sc;
  shift[c] = beta[c] - mu * sc;
}

// ---------------------------------------------------------------------------
// Layer 1: A = relu(BN(Z0)) loaded once into registers, weights in LDS,
// loop all 7 n-tiles. Store Z1 + stats partials.
// ---------------------------------------------------------------------------
__global__ __launch_bounds__(256) void gbdt_gemm_mid(
    const float* __restrict__ Zin, const float* __restrict__ w,
    const float* __restrict__ b, const float* __restrict__ scale,
    const float* __restrict__ shift, float* __restrict__ Zout,
    float* __restrict__ psum, float* __restrict__ psq)
{
  __shared__ alignas(16) float wtT[kNT01 * 16][kL0 + 4];  // 112x104
  __shared__ alignas(16) float sc_s[kL0 + 4];
  __shared__ alignas(16) float sh_s[kL0 + 4];
  __shared__ float bias_s[kNT01 * 16];
  __shared__ float wsum[kWaves][16];
  __shared__ float wsq[kWaves][16];

  const int g    = blockIdx.y;
  const int tid  = threadIdx.x;
  const int wave = tid >> 5;
  const int lane = tid & 31;
  const int half = lane >> 4;
  const int r    = lane & 15;
  const int m0   = (blockIdx.x * kWaves + wave) * 16;

  for (int i = tid; i < kL0; i += 256) {
    sc_s[i] = scale[g * kL0 + i];
    sh_s[i] = shift[g * kL0 + i];
  }
  for (int i = tid; i < kL0 * kNT01 * 16; i += 256) {
    const int k = i / (kNT01 * 16), n = i % (kNT01 * 16);
    wtT[n][k] = (n < kL1) ? w[((size_t)g * kL0 + k) * kL1 + n] : 0.0f;
  }
  if (tid < kNT01 * 16) bias_s[tid] = (tid < kL1) ? b[g * kL1 + tid] : 0.0f;
  __syncthreads();

  // A fragments: BN + ReLU applied once here (50 VGPRs/lane)
  const float* __restrict__ zrow = Zin + ((size_t)g * kB + m0 + r) * kL0;
  v2f afrag[kL0 / 4];
#pragma unroll
  for (int kb = 0; kb < kL0 / 4; ++kb) {
    const int k   = kb * 4 + half * 2;
    const v2f zv  = *(const v2f*)&zrow[k];
    const v2f scv = *(const v2f*)&sc_s[k];
    const v2f shv = *(const v2f*)&sh_s[k];
    afrag[kb].x = fmaxf(fmaf(zv.x, scv.x, shv.x), 0.0f);
    afrag[kb].y = fmaxf(fmaf(zv.y, scv.y, shv.y), 0.0f);
  }

  for (int nt = 0; nt < kNT01; ++nt) {
    const int n0 = nt * 16;
    v8f acc = {};
#pragma unroll
    for (int kb = 0; kb < kL0 / 4; ++kb) {
      const int k = kb * 4 + half * 2;
      const v2f bb = *(const v2f*)&wtT[n0 + r][k];
      acc = wmma_f32(afrag[kb], bb, acc);
    }
    const int n = n0 + r;
    float s = 0.0f, q = 0.0f;
    if (n < kL1) {
      const float bia = bias_s[n];
      float* __restrict__ zp = Zout + ((size_t)g * kB + m0 + half * 8) * kL1 + n;
#pragma unroll
      for (int j = 0; j < 8; ++j) {
        const float z = acc[j] + bia;
        zp[j * kL1] = z;
        s += z;
        q += z * z;
      }
    }
    s += __shfl_xor(s, 16);
    q += __shfl_xor(q, 16);
    if (half == 0) { wsum[wave][r] = s; wsq[wave][r] = q; }
    __syncthreads();
    if (tid < 16 && (n0 + tid) < kL1) {
      float ts = 0.0f, tq = 0.0f;
#pragma unroll
      for (int w2 = 0; w2 < kWaves; ++w2) { ts += wsum[w2][tid]; tq += wsq[w2][tid]; }
      const size_t c = (size_t)(g * kL1 + n0 + tid) * kMBlocks + blockIdx.x;
      psum[c] = ts;
      psq[c]  = tq;
    }
    __syncthreads();
  }
}

// ---------------------------------------------------------------------------
// Layer 2: A = relu(BN(Z1)) in registers, 4 n-tiles. Writes pred and
// per-(row,group) partial of pred . out_weight (shfl-tree reduced).
// ---------------------------------------------------------------------------
__global__ __launch_bounds__(256) void gbdt_gemm_last(
    const float* __restrict__ Zin, const float* __restrict__ w,
    const float* __restrict__ b, const float* __restrict__ scale,
    const float* __restrict__ shift, const float* __restrict__ wout,
    float* __restrict__ pred, float* __restrict__ rowpart)
{
  __shared__ alignas(16) float wtT[kNT2 * 16][kL1 + 4];   // 64x104
  __shared__ alignas(16) float sc_s[kL1 + 4];
  __shared__ alignas(16) float sh_s[kL1 + 4];
  __shared__ float bias_s[kNT2 * 16];
  __shared__ float wout_s[kNT2 * 16];

  const int g    = blockIdx.y;
  const int tid  = threadIdx.x;
  const int wave = tid >> 5;
  const int lane = tid & 31;
  const int half = lane >> 4;
  const int r    = lane & 15;
  const int m0   = (blockIdx.x * kWaves + wave) * 16;

  for (int i = tid; i < kL1; i += 256) {
    sc_s[i] = scale[g * kL1 + i];
    sh_s[i] = shift[g * kL1 + i];
  }
  for (int i = tid; i < kL1 * kNT2 * 16; i += 256) {
    const int k = i / (kNT2 * 16), n = i % (kNT2 * 16);
    wtT[n][k] = (n < kL2) ? w[((size_t)g * kL1 + k) * kL2 + n] : 0.0f;
  }
  if (tid < kNT2 * 16) {
    bias_s[tid] = (tid < kL2) ? b[g * kL2 + tid] : 0.0f;
    wout_s[tid] = (tid < kL2) ? wout[g * kL2 + tid] : 0.0f;
  }
  __syncthreads();

  const float* __restrict__ zrow = Zin + ((size_t)g * kB + m0 + r) * kL1;
  v2f afrag[kL1 / 4];
#pragma unroll
  for (int kb = 0; kb < kL1 / 4; ++kb) {
    const int k   = kb * 4 + half * 2;
    const v2f zv  = *(const v2f*)&zrow[k];
    const v2f scv = *(const v2f*)&sc_s[k];
    const v2f shv = *(const v2f*)&sh_s[k];
    afrag[kb].x = fmaxf(fmaf(zv.x, scv.x, shv.x), 0.0f);
    afrag[kb].y = fmaxf(fmaf(zv.y, scv.y, shv.y), 0.0f);
  }

  float dotj[8] = {0.f, 0.f, 0.f, 0.f, 0.f, 0.f, 0.f, 0.f};
  for (int nt = 0; nt < kNT2; ++nt) {
    const int n0 = nt * 16;
    v8f acc = {};
#pragma unroll
    for (int kb = 0; kb < kL1 / 4; ++kb) {
      const int k = kb * 4 + half * 2;
      const v2f bb = *(const v2f*)&wtT[n0 + r][k];
      acc = wmma_f32(afrag[kb], bb, acc);
    }
    const int n = n0 + r;
    if (n < kL2) {
      const float bia = bias_s[n];
      const float wn  = wout_s[n];
      float* __restrict__ pp =
          pred + (size_t)(m0 + half * 8) * (kG * kL2) + (size_t)g * kL2 + n;
#pragma unroll
      for (int j = 0; j < 8; ++j) {
        const float p = acc[j] + bia;
        pp[(size_t)j * (kG * kL2)] = p;
        dotj[j] = fmaf(p, wn, dotj[j]);
      }
    }
  }
  // Reduce each row's dot over the 16 N-lanes (fixed shfl tree)
#pragma unroll
  for (int j = 0; j < 8; ++j) {
    float v = dotj[j];
    v += __shfl_xor(v, 1);
    v += __shfl_xor(v, 2);
    v += __shfl_xor(v, 4);
    v += __shfl_xor(v, 8);
    if (r == 0) rowpart[(size_t)(m0 + half * 8 + j) * kG + g] = v;
  }
}

// ---------------------------------------------------------------------------
// out[b] = sigmoid(sum_g rowpart[b][g] + out_bias)
// ---------------------------------------------------------------------------
__global__ __launch_bounds__(256) void gbdt_final(
    const float* __restrict__ rowpart, const float* __restrict__ obias,
    float* __restrict__ out)
{
  const int row = blockIdx.x * 256 + threadIdx.x;
  if (row >= kB) return;
  const float* __restrict__ rp = rowpart + (size_t)row * kG;
  float t = 0.0f;
  for (int g = 0; g < kG; ++g) t += rp[g];
  t += obias[0];
  out[row] = 1.0f / (1.0f + __expf(-t));
}

// ---------------------------------------------------------------------------
extern "C" void kernel_launch(void* const* d_in, const int* in_sizes, int n_in,
                              void* d_out, int out_size, void* d_ws, size_t ws_size,
                              hipStream_t stream) {
  (void)in_sizes; (void)n_in; (void)out_size; (void)ws_size;

  const float* x   = (const float*)d_in[0];
  const int*   uf  = (const int*)d_in[1];
  const float* w0  = (const float*)d_in[2];
  const float* b0  = (const float*)d_in[3];
  const float* w1  = (const float*)d_in[4];
  const float* b1  = (const float*)d_in[5];
  const float* w2  = (const float*)d_in[6];
  const float* b2  = (const float*)d_in[7];
  const float* g0  = (const float*)d_in[8];
  const float* be0 = (const float*)d_in[9];
  const float* g1  = (const float*)d_in[10];
  const float* be1 = (const float*)d_in[11];
  const float* ow  = (const float*)d_in[12];
  const float* ob  = (const float*)d_in[13];

  // Workspace layout (floats). Everything rewritten every launch -> no init.
  float* ws = (float*)d_ws;
  constexpr size_t kNCh  = (size_t)kG * kL0;               // 10000
  constexpr size_t kPart = kNCh * kMBlocks;                // 640000
  float* sc0     = ws;
  float* sh0     = sc0 + kNCh;
  float* sc1     = sh0 + kNCh;
  float* sh1     = sc1 + kNCh;
  float* psum0   = sh1 + kNCh;
  float* psq0    = psum0 + kPart;
  float* psum1   = psq0 + kPart;
  float* psq1    = psum1 + kPart;
  float* rowpart = psq1 + kPart;                           // [B][G]
  float* Z0      = rowpart + (size_t)kB * kG;              // [G][B][100]
  float* Z1      = Z0 + (size_t)kG * kB * kL0;             // [G][B][100]

  float* out_sig = (float*)d_out;           // [B, 1]
  float* pred    = out_sig + kB;            // [B, G*50]

  const dim3 blk(256);
  gbdt_gemm0<<<dim3(kMBlocks, kG), blk, 0, stream>>>(
      x, uf, w0, b0, Z0, psum0, psq0);
  gbdt_bnprep<<<(kG * kL0 + 255) / 256, blk, 0, stream>>>(
      psum0, psq0, g0, be0, sc0, sh0);
  gbdt_gemm_mid<<<dim3(kMBlocks, kG), blk, 0, stream>>>(
      Z0, w1, b1, sc0, sh0, Z1, psum1, psq1);
  gbdt_bnprep<<<(kG * kL1 + 255) / 256, blk, 0, stream>>>(
      psum1, psq1, g1, be1, sc1, sh1);
  gbdt_gemm_last<<<dim3(kMBlocks, kG), blk, 0, stream>>>(
      Z1, w2, b2, sc1, sh1, ow, pred, rowpart);
  gbdt_final<<<(kB + 255) / 256, blk, 0, stream>>>(rowpart, ob, out_sig);
}